// Router_84868553769173
// MI455X (gfx1250) — compile-verified
//
#include <hip/hip_runtime.h>
#include <hip/hip_bf16.h>

typedef __attribute__((ext_vector_type(2))) float v2f;
typedef __attribute__((ext_vector_type(8))) float v8f;

#define N_EMBD 2048
#define ROUTE  64
#define ROWS_PER_WAVE 32           // 2 M-tiles of 16
#define WAVES_PER_BLOCK 4
#define THREADS_PER_BLOCK (WAVES_PER_BLOCK * 32)
#define ROWS_PER_BLOCK (ROWS_PER_WAVE * WAVES_PER_BLOCK)   // 128
#define NEG_INF (-3.402823466e+38f)

__global__ __launch_bounds__(THREADS_PER_BLOCK) void router_wmma_f32_kernel(
    const float* __restrict__ x,      // [32768, 2048]
    const float* __restrict__ W,      // [64, 2048]
    int*   __restrict__ idx_out,      // [32768, 2]
    float* __restrict__ w_out,        // [32768, 2]
    float* __restrict__ logits_out)   // [32768, 64]
{
    // Per-wave 32x64 logit tile staged for top-k; row stride 65 -> conflict-free
    __shared__ float lds[WAVES_PER_BLOCK][ROWS_PER_WAVE][ROUTE + 1];

    const int tid   = threadIdx.x;
    const int wave  = tid >> 5;
    const int lane  = tid & 31;
    const int lhalf = lane >> 4;   // 0: lanes 0-15, 1: lanes 16-31
    const int l15   = lane & 15;
    const int rbase = blockIdx.x * ROWS_PER_BLOCK + wave * ROWS_PER_WAVE;

    // A fragments: M-tile m covers rows rbase + 16*m + (lane&15); K offset 0/2 by lane half.
    // Second tile reached via constant +16*N_EMBD element offset (fits 24-bit IOFFSET).
    const float* xrow = x + (size_t)(rbase + l15) * N_EMBD + (lhalf << 1);
    // B fragments: expert n = 16*t + (lane&15), same K split (logits = x.W^T)
    const float* wrow = W + (size_t)l15 * N_EMBD + (lhalf << 1);

    v8f acc[2][4];
    #pragma unroll
    for (int m = 0; m < 2; ++m)
        #pragma unroll
        for (int t = 0; t < 4; ++t)
            acc[m][t] = (v8f){};

    #pragma unroll 4
    for (int k = 0; k < N_EMBD; k += 4) {
        v2f a0 = *(const v2f*)(xrow + k);
        v2f a1 = *(const v2f*)(xrow + 16 * N_EMBD + k);
        v2f b0 = *(const v2f*)(wrow + k);
        v2f b1 = *(const v2f*)(wrow + 16 * N_EMBD + k);
        v2f b2 = *(const v2f*)(wrow + 32 * N_EMBD + k);
        v2f b3 = *(const v2f*)(wrow + 48 * N_EMBD + k);
        // D = A(16x4) * B(4x16) + C ; fp32 WMMA, 8 per 4-K step
        acc[0][0] = __builtin_amdgcn_wmma_f32_16x16x4_f32(false, a0, false, b0, (short)0, acc[0][0], false, false);
        acc[0][1] = __builtin_amdgcn_wmma_f32_16x16x4_f32(false, a0, false, b1, (short)0, acc[0][1], false, false);
        acc[0][2] = __builtin_amdgcn_wmma_f32_16x16x4_f32(false, a0, false, b2, (short)0, acc[0][2], false, false);
        acc[0][3] = __builtin_amdgcn_wmma_f32_16x16x4_f32(false, a0, false, b3, (short)0, acc[0][3], false, false);
        acc[1][0] = __builtin_amdgcn_wmma_f32_16x16x4_f32(false, a1, false, b0, (short)0, acc[1][0], false, false);
        acc[1][1] = __builtin_amdgcn_wmma_f32_16x16x4_f32(false, a1, false, b1, (short)0, acc[1][1], false, false);
        acc[1][2] = __builtin_amdgcn_wmma_f32_16x16x4_f32(false, a1, false, b2, (short)0, acc[1][2], false, false);
        acc[1][3] = __builtin_amdgcn_wmma_f32_16x16x4_f32(false, a1, false, b3, (short)0, acc[1][3], false, false);
    }

    // C/D layout: VGPR g holds row (16*m + g + 8*lhalf), col = 16*t + (lane&15).
    // Emit logits to global and stage the 32x64 tile in LDS for the top-k pass.
    #pragma unroll
    for (int m = 0; m < 2; ++m) {
        #pragma unroll
        for (int t = 0; t < 4; ++t) {
            const int e = (t << 4) + l15;
            #pragma unroll
            for (int g = 0; g < 8; ++g) {
                const int row = (m << 4) + g + (lhalf << 3);
                const float v = acc[m][t][g];
                lds[wave][row][e] = v;
                logits_out[(size_t)(rbase + row) * ROUTE + e] = v;
            }
        }
    }

    // Same-wave LDS ops are in-order; wait for stores before cross-lane reads.
    asm volatile("s_wait_dscnt 0" ::: "memory");

    // All 32 lanes active: one row per lane.
    {
        const int row = lane;
        float v0 = NEG_INF, v1 = NEG_INF;
        int   i0 = 0,       i1 = 0;
        // Stable top-2: strict > keeps the earliest index on ties,
        // matching stable argsort of -scores in the reference.
        #pragma unroll 8
        for (int e = 0; e < ROUTE; ++e) {
            const float v = lds[wave][row][e];
            if (v > v0)      { v1 = v0; i1 = i0; v0 = v; i0 = e; }
            else if (v > v1) { v1 = v;  i1 = e; }
        }
        // softmax over [v0, v1] with v0 >= v1
        const float ee  = __expf(v1 - v0);
        const float inv = 1.0f / (1.0f + ee);
        const size_t r  = (size_t)(rbase + row);
        idx_out[2 * r]     = i0;
        idx_out[2 * r + 1] = i1;
        w_out[2 * r]       = inv;
        w_out[2 * r + 1]   = ee * inv;
    }
}

extern "C" void kernel_launch(void* const* d_in, const int* in_sizes, int n_in,
                              void* d_out, int out_size, void* d_ws, size_t ws_size,
                              hipStream_t stream) {
    (void)in_sizes; (void)n_in; (void)d_ws; (void)ws_size; (void)out_size;

    const float* x = (const float*)d_in[0];   // 32768 x 2048 fp32
    const float* W = (const float*)d_in[1];   // 64 x 2048 fp32

    const int T = 32768;
    // d_out: [indices i32 (T*2)] [weights f32 (T*2)] [logits f32 (T*64)]
    int*   idx_out    = (int*)d_out;
    float* w_out      = (float*)d_out + (size_t)T * 2;
    float* logits_out = (float*)d_out + (size_t)T * 4;

    const int blocks = T / ROWS_PER_BLOCK;    // 32768 / 128 = 256
    router_wmma_f32_kernel<<<blocks, THREADS_PER_BLOCK, 0, stream>>>(x, W, idx_out, w_out, logits_out);
}